// GetAffs_83416854823610
// MI455X (gfx1250) — compile-verified
//
#include <hip/hip_runtime.h>
#include <cstdint>

// ---------------------------------------------------------------------------
// GetAffs on MI455X (gfx1250): memory-bound 8-offset label-affinity stencil.
//   in : seg  [1,1,4096,4096] f32 (integer labels as float)
//   out: aff  [8,4096,4096]   f32, plane k: (shift(seg, off_k) == seg) ? 1 : 0
//        with zero padding outside the image (matches reference _shift()).
// Roofline: 64MB reads + 512MB writes @ 23.3TB/s -> ~25us; no matmul content,
// so the CDNA5 win is the data path: TDM tensor_load_to_lds for the input tile
// (TENSORcnt + s_wait_tensorcnt) and non-temporal b128 stores for the output.
// ---------------------------------------------------------------------------

#define W        4096
#define H        4096
#define TILE     64
#define HALO     8                 // largest negative offset
#define LW       (TILE + HALO)     // 72 (LDS row = TDM tile_dim0, contiguous)
#define LH       (TILE + HALO)     // 72
#define NTHREADS 256

typedef __attribute__((ext_vector_type(4))) float        v4f;
typedef __attribute__((ext_vector_type(4))) unsigned int v4u_t;
typedef __attribute__((ext_vector_type(8))) int          v8i_t;
typedef __attribute__((ext_vector_type(4))) int          v4i_t;

#if __has_builtin(__builtin_amdgcn_tensor_load_to_lds)
#define HAVE_TDM 1
#else
#define HAVE_TDM 0
#endif

#if HAVE_TDM
// Issue one TDM DMA: 72x72 f32 tile starting at gsrc (row stride 4096 elems)
// into LDS offset 0 (single static __shared__ array). D# layout per CDNA5 ISA
// ch. 8 (group0: count/lds_addr/global_addr/type; group1: data_size, dims,
// tile dims, stride). Tensor dims set huge: interior tiles are never OOB.
__device__ __forceinline__ void tdm_load_tile(const float* gsrc) {
  const uint64_t ga = (uint64_t)(uintptr_t)gsrc;
  v4u_t g0;
  g0.x = 1u;                                              // count=1, user desc
  g0.y = 0u;                                              // lds_addr = 0
  g0.z = (uint32_t)(ga & 0xFFFFFFFFu);                    // global_addr[31:0]
  g0.w = (uint32_t)((ga >> 32) & 0x01FFFFFFu) | (2u << 30); // addr[56:32]|type=2

  const uint32_t TD = 1u << 20;                           // big tensor dims
  v8i_t g1;
  g1[0] = (int)(2u << 16);                                // data_size=4B, mask=0
  g1[1] = (int)((TD & 0xFFFFu) << 16);                    // tensor_dim0[15:0]
  g1[2] = (int)((TD >> 16) | ((TD & 0xFFFFu) << 16));     // dim0 hi | dim1 lo
  g1[3] = (int)((TD >> 16) | ((uint32_t)LW << 16));       // dim1 hi | tile_dim0=72
  g1[4] = (int)LH;                                        // tile_dim1=72, dim2=0
  g1[5] = (int)W;                                         // dim0_stride = 4096
  g1[6] = 0;
  g1[7] = 0;

  v4i_t gz = {0, 0, 0, 0};
#if defined(__clang_major__) && __clang_major__ >= 23
  v8i_t gz8 = {0, 0, 0, 0, 0, 0, 0, 0};
  __builtin_amdgcn_tensor_load_to_lds(g0, g1, gz, gz, gz8, 0);
#else
  __builtin_amdgcn_tensor_load_to_lds(g0, g1, gz, gz, 0);
#endif
  __builtin_amdgcn_s_wait_tensorcnt(0);                   // TENSORcnt -> 0
}
#endif

__global__ __launch_bounds__(NTHREADS)
void GetAffs_83416854823610_kernel(const float* __restrict__ seg,
                                   float* __restrict__ out) {
  __shared__ float tile[LH * LW];                         // 20.25 KB of 320 KB

  const int bx = blockIdx.x, by = blockIdx.y;
  const int gx0 = bx * TILE, gy0 = by * TILE;
  const int tid = (int)threadIdx.x;

#if HAVE_TDM
  if (bx > 0 && by > 0) {
    // Interior tile: whole 72x72 window (incl. top/left halo) is in-bounds.
    // One wave issues the async tensor DMA and waits on TENSORcnt.
    if (tid < 32) {
      tdm_load_tile(seg + (size_t)(gy0 - HALO) * W + (gx0 - HALO));
    }
  } else
#endif
  {
    // Boundary tile (or no TDM builtin): cooperative fill, explicit zero pad
    // above/left of the image — exactly the reference's zero-padded shift.
    for (int i = tid; i < LH * LW; i += NTHREADS) {
      const int r = i / LW, c = i % LW;
      const int gy = gy0 - HALO + r, gx = gx0 - HALO + c;
      float v = 0.0f;
      if (gy >= 0 && gx >= 0) v = seg[(size_t)gy * W + gx];
      tile[i] = v;
    }
  }
  __syncthreads();   // uniform branch per block -> barrier is safe

  // Each thread: 4 rows x 4 consecutive cols. Lanes 0..15 of a wave cover one
  // 64-wide row as 16B chunks -> fully coalesced 256B store bursts.
  const int tx4 = (tid & 15) << 2;        // 0,4,...,60
  const int rg  = tid >> 4;               // 0..15
  const size_t HW = (size_t)H * W;

#pragma unroll
  for (int rr = 0; rr < 4; ++rr) {
    const int row  = rg + 16 * rr;        // 0..63
    const int lrow = HALO + row;

    // Sliding 12-float window: LDS cols [tx4 .. tx4+11] of the center row
    // covers center (cols +8..+11) and col offsets -1,-2,-4,-8.
    const float* rowp = &tile[lrow * LW + tx4];
    float sv[12];
#pragma unroll
    for (int i = 0; i < 12; ++i) sv[i] = rowp[i];

    // Row-offset neighbors, 16B-aligned v4f LDS loads (LW=72, tx4+8 ≡ 0 mod 4).
    const v4f r1 = *(const v4f*)&tile[(lrow - 1) * LW + tx4 + 8];
    const v4f r2 = *(const v4f*)&tile[(lrow - 2) * LW + tx4 + 8];
    const v4f r4 = *(const v4f*)&tile[(lrow - 4) * LW + tx4 + 8];
    const v4f r8 = *(const v4f*)&tile[(lrow - 8) * LW + tx4 + 8];

    v4f a0, a1, a2, a3, a4, a5, a6, a7;
#pragma unroll
    for (int p = 0; p < 4; ++p) {
      const float c = sv[8 + p];                       // center pixel
      a0[p] = (r1[p]     == c) ? 1.0f : 0.0f;          // (-1, 0)
      a1[p] = (sv[7 + p] == c) ? 1.0f : 0.0f;          // ( 0,-1)
      a2[p] = (r2[p]     == c) ? 1.0f : 0.0f;          // (-2, 0)
      a3[p] = (sv[6 + p] == c) ? 1.0f : 0.0f;          // ( 0,-2)
      a4[p] = (r4[p]     == c) ? 1.0f : 0.0f;          // (-4, 0)
      a5[p] = (sv[4 + p] == c) ? 1.0f : 0.0f;          // ( 0,-4)
      a6[p] = (r8[p]     == c) ? 1.0f : 0.0f;          // (-8, 0)
      a7[p] = (sv[0 + p] == c) ? 1.0f : 0.0f;          // ( 0,-8)
    }

    // Streamed output (written once, never re-read): non-temporal b128 stores
    // keep the 512MB output from evicting reusable input rows in L2.
    float* obase = out + (size_t)(gy0 + row) * W + (size_t)(gx0 + tx4);
    __builtin_nontemporal_store(a0, (v4f*)(obase + 0 * HW));
    __builtin_nontemporal_store(a1, (v4f*)(obase + 1 * HW));
    __builtin_nontemporal_store(a2, (v4f*)(obase + 2 * HW));
    __builtin_nontemporal_store(a3, (v4f*)(obase + 3 * HW));
    __builtin_nontemporal_store(a4, (v4f*)(obase + 4 * HW));
    __builtin_nontemporal_store(a5, (v4f*)(obase + 5 * HW));
    __builtin_nontemporal_store(a6, (v4f*)(obase + 6 * HW));
    __builtin_nontemporal_store(a7, (v4f*)(obase + 7 * HW));
  }
}

extern "C" void kernel_launch(void* const* d_in, const int* in_sizes, int n_in,
                              void* d_out, int out_size, void* d_ws, size_t ws_size,
                              hipStream_t stream) {
  (void)in_sizes; (void)n_in; (void)out_size; (void)d_ws; (void)ws_size;
  const float* seg = (const float*)d_in[0];
  float* out = (float*)d_out;
  dim3 grid(W / TILE, H / TILE);     // 64 x 64 tiles
  GetAffs_83416854823610_kernel<<<grid, NTHREADS, 0, stream>>>(seg, out);
}